// RSDPF_84765474554407
// MI455X (gfx1250) — compile-verified
//
#include <hip/hip_runtime.h>
#include <hip/hip_bf16.h>

// Problem constants (fixed by the reference)
#define BATCH    64
#define NP       2048
#define TLEN     100
#define NMODES   8
#define NTHREADS 256
#define PPT      (NP / NTHREADS)   // 8 particles per thread

typedef __attribute__((ext_vector_type(2))) float v2f;
typedef __attribute__((ext_vector_type(8))) float v8f;

// ---------------- stateless counter-based RNG (deterministic, graph-safe) ---
__device__ __forceinline__ uint32_t hash_u32(uint32_t x) {
    x ^= x >> 16; x *= 0x7feb352du;
    x ^= x >> 15; x *= 0x846ca68bu;
    x ^= x >> 16;
    return x;
}
// short finalizer for deriving substreams from one well-mixed base
__device__ __forceinline__ uint32_t mix_u32(uint32_t x) {
    x *= 0x9E3779B9u; x ^= x >> 16; x *= 0x85EBCA77u; x ^= x >> 13;
    return x;
}
__device__ __forceinline__ uint32_t rng3(uint32_t a, uint32_t b, uint32_t c) {
    return hash_u32(hash_u32(a * 0x9E3779B9u + b) ^ (c * 0x85EBCA77u) ^ 0x27220A95u);
}
__device__ __forceinline__ float u01(uint32_t h)  {   // [0, 1)
    return (float)h * 2.3283064365386963e-10f;
}
__device__ __forceinline__ float u01o(uint32_t h) {   // (0, 1) strictly positive
    return ((float)h + 0.5f) * 2.3283064365386963e-10f;
}

__global__ __launch_bounds__(NTHREADS)
void rsdpf_filter_kernel(const float* __restrict__ o,        // [B, 1, T]
                         const float* __restrict__ P,        // [NM, NM]
                         const float* __restrict__ co_A,     // [NM]
                         const float* __restrict__ co_B,     // [NM]
                         const float* __restrict__ co_C,     // [NM]
                         const float* __restrict__ co_D,     // [NM]
                         const float* __restrict__ sigma_u,  // [1]
                         const float* __restrict__ sigma_v,  // [1]
                         float* __restrict__ out)            // [B, T]
{
    // Entire per-batch particle state lives in LDS (~28.5 KB of 320 KB/WGP).
    __shared__ float s_lds[NP];
    __shared__ int   m_lds[NP];
    __shared__ float cumw[NP];
    __shared__ float scan_in[NTHREADS];
    __shared__ float scan_out[NTHREADS];
    __shared__ float red_in[NTHREADS];
    __shared__ float obs[TLEN + 4];
    __shared__ float cumP[NMODES][NMODES];
    __shared__ float cA[NMODES], cB[NMODES], cC[NMODES], cD[NMODES];
    __shared__ float s_misc[4];   // [0]=sigma_u [1]=sigma_v [2]=block reduction

    const int tid = threadIdx.x;
    const int gb  = blockIdx.x;

    // ---------------- init: tiny input staging -----------------------------
    __builtin_prefetch(o + gb * TLEN, 0, 1);     // global_prefetch_b8
    for (int i = tid; i < TLEN; i += NTHREADS) obs[i] = o[gb * TLEN + i];
    if (tid < NMODES) {
        cA[tid] = co_A[tid]; cB[tid] = co_B[tid];
        cC[tid] = co_C[tid]; cD[tid] = co_D[tid];
        float acc = 0.0f;
        for (int k = 0; k < NMODES; ++k) { acc += P[tid * NMODES + k]; cumP[tid][k] = acc; }
    }
    if (tid == 0) { s_misc[0] = sigma_u[0]; s_misc[1] = sigma_v[0]; }

    // ---------------- init particles ---------------------------------------
    float part_s = 0.0f;
    #pragma unroll
    for (int j = 0; j < PPT; ++j) {
        const int p = tid * PPT + j;
        const uint32_t h0 = rng3((uint32_t)gb, 0u, (uint32_t)p);
        const int   m0 = (int)(h0 & (NMODES - 1));
        const float s0 = u01(mix_u32(h0 ^ 0x51AFE213u)) - 0.5f;
        m_lds[p] = m0; s_lds[p] = s0;
        part_s += s0;
    }
    red_in[tid] = part_s;
    __syncthreads();

    const float su     = s_misc[0];
    const float sv     = s_misc[1];
    const float inv_sv = 1.0f / sv;
    const float norm_c = 0.3989422804014327f * inv_sv;  // 1/(sv*sqrt(2*pi))
    const float invNp  = 1.0f / (float)NP;

    // t = 0 posterior mean = mean(s0) (weights are uniform 1/Np)
    if (tid >= 32 && tid < 64) {
        const int lane = tid - 32;
        float v = 0.0f;
        #pragma unroll
        for (int k = 0; k < 8; ++k) v += red_in[lane + 32 * k];
        #pragma unroll
        for (int d = 16; d > 0; d >>= 1) v += __shfl_down(v, d, 32);
        if (lane == 0) s_misc[2] = v;
    }
    __syncthreads();
    if (tid == 0) out[gb * TLEN + 0] = s_misc[2] * invNp;

    // ---------------- sequential filtering steps ---------------------------
    for (int t = 1; t < TLEN; ++t) {
        const float ot = obs[t];
        float wloc_sum = 0.0f, dotp = 0.0f;
        float winc[PPT];

        #pragma unroll
        for (int j = 0; j < PPT; ++j) {
            const int   p    = tid * PPT + j;
            const int   mold = m_lds[p];
            const float sold = s_lds[p];
            // one well-mixed base hash per (batch, step, particle), 3 substreams
            const uint32_t hb = rng3((uint32_t)gb, (uint32_t)t, (uint32_t)p);
            // Markov regime sampling (inverse-CDF on precomputed row cumsum)
            const float u = u01(hb);
            int mnew = 0;
            #pragma unroll
            for (int k = 0; k < NMODES; ++k) mnew += (u >= cumP[mold][k]) ? 1 : 0;
            if (mnew > NMODES - 1) mnew = NMODES - 1;
            // Gaussian process noise via Box-Muller (two derived substreams)
            const float u1  = u01o(mix_u32(hb ^ 0xC2B2AE35u));
            const float u2  = u01 (mix_u32(hb ^ 0x165667B1u));
            const float nrm = sqrtf(-2.0f * __logf(u1)) * __cosf(6.283185307179586f * u2);
            float snew = cA[mnew] * sold + cB[mnew] + nrm * su;
            snew = (snew == 0.0f) ? 1.0e-4f : snew;
            // bootstrap weight (Gaussian obs likelihood), w_re == 1/Np always
            const float mean = cC[mnew] * sqrtf(fabsf(snew)) + cD[mnew];
            const float z    = (ot - mean) * inv_sv;
            const float w    = norm_c * __expf(-0.5f * z * z) * invNp;
            s_lds[p] = snew; m_lds[p] = mnew;
            wloc_sum += w; winc[j] = wloc_sum;
            dotp += w * snew;
        }
        scan_in[tid] = wloc_sum;   // 256 per-thread partial sums
        red_in[tid]  = dotp;       // 256 partial dot products
        __syncthreads();

        if (tid < 32) {
            // ---- wave 0: exact f32 scan of 256 partials via 4x WMMA -------
            // Value v = n*16 + k laid out as 16 columns (n) x 16 rows (k).
            // D = sum_i A_i x B_i with A_i = columns [4i,4i+4) of the 16x16
            // lower-triangular ones matrix  =>  D(m,n) = sum_{k<=m} X(k,n).
            const int  n  = tid & 15;        // column (and A-row m; same L%16)
            const bool hi = tid >= 16;
            v8f c = {};
            #pragma unroll
            for (int i = 0; i < 4; ++i) {
                v2f a, b;
                const int ka = 4 * i + (hi ? 2 : 0);   // A cols / B rows this lane holds
                a.x = ((ka    ) <= n) ? 1.0f : 0.0f;   // A(m=n, k=ka)
                a.y = ((ka + 1) <= n) ? 1.0f : 0.0f;   // A(m=n, k=ka+1)
                b.x = scan_in[n * 16 + ka];            // B(k=ka,   col n)
                b.y = scan_in[n * 16 + ka + 1];        // B(k=ka+1, col n)
                c = __builtin_amdgcn_wmma_f32_16x16x4_f32(
                        /*neg_a=*/false, a, /*neg_b=*/false, b,
                        /*c_mod=*/(short)0, c, /*reuse_a=*/false, /*reuse_b=*/false);
            }
            // Cross-column exclusive offsets. Column totals D(15,n) live in
            // c[7] of lanes 16..31 (C/D layout: VGPR7 hi-half == M=15).
            // Log-depth inclusive scan across lanes 16..31, then one
            // bpermute broadcast of the exclusive value to both halves.
            const float tot = c[7];
            float inc = tot;
            #pragma unroll
            for (int d = 1; d < 16; d <<= 1) {
                const float tv = __shfl_up(inc, d, 32);
                inc += (tid - d >= 16) ? tv : 0.0f;
            }
            const float exc = inc - tot;               // valid in lanes 16..31
            const float off = __shfl(exc, 16 + n, 32); // off[n] for this column
            #pragma unroll
            for (int j = 0; j < 8; ++j) {
                const int m = j + (hi ? 8 : 0);
                scan_out[n * 16 + m] = c[j] + off;     // global inclusive scan
            }
        } else if (tid < 64) {
            // ---- wave 1 (concurrent): reduce sum(w*s) ---------------------
            const int lane = tid - 32;
            float v = 0.0f;
            #pragma unroll
            for (int k = 0; k < 8; ++k) v += red_in[lane + 32 * k];
            #pragma unroll
            for (int d = 16; d > 0; d >>= 1) v += __shfl_down(v, d, 32);
            if (lane == 0) s_misc[2] = v;
        }
        __syncthreads();

        const float incl = scan_out[tid];
        const float excl = incl - wloc_sum;
        const float W    = scan_out[NTHREADS - 1];     // total weight
        #pragma unroll
        for (int j = 0; j < PPT; ++j) cumw[tid * PPT + j] = excl + winc[j];
        if (tid == 0) out[gb * TLEN + t] = s_misc[2] / W;   // (sum w*s)/W
        __syncthreads();

        // ---- systematic resampling --------------------------------------
        // This thread's targets are increasing, so result indices are
        // monotone: gallop (exponential search) from the previous index,
        // then a short binary search in the bracketed range.
        const float u0 = u01(rng3((uint32_t)gb, 0xBEEF0000u + (uint32_t)t, 0x123457u));
        const float Wn = W * invNp;
        float sg[PPT]; int mg[PPT];
        int prev = 0;
        #pragma unroll
        for (int j = 0; j < PPT; ++j) {
            const int   p      = tid * PPT + j;
            const float target = ((float)p + u0) * Wn;
            int idx;
            if (cumw[prev] >= target) {
                idx = prev;                     // still inside previous bucket
            } else {
                int lo = prev, step = 1;        // gallop: cumw[lo] < target
                while (lo + step < NP && cumw[lo + step] < target) {
                    lo += step; step <<= 1;
                }
                int a = lo + 1;
                int b = (lo + step < NP) ? (lo + step) : NP;
                while (a < b) {                 // lower_bound in (lo, b]
                    const int mid = (a + b) >> 1;
                    if (cumw[mid] < target) a = mid + 1; else b = mid;
                }
                idx = (a < NP) ? a : (NP - 1);
            }
            prev = idx;
            sg[j] = s_lds[idx]; mg[j] = m_lds[idx];
        }
        __syncthreads();                    // all gathers done before scatter
        #pragma unroll
        for (int j = 0; j < PPT; ++j) {
            const int p = tid * PPT + j;
            s_lds[p] = sg[j]; m_lds[p] = mg[j];
        }
        __syncthreads();
    }
}

extern "C" void kernel_launch(void* const* d_in, const int* in_sizes, int n_in,
                              void* d_out, int out_size, void* d_ws, size_t ws_size,
                              hipStream_t stream) {
    // setup_inputs order: o, P, beta(unused), co_A, co_B, co_C, co_D,
    //                     sigma_u, sigma_v, N_p(compile-time const here)
    const float* o       = (const float*)d_in[0];
    const float* P       = (const float*)d_in[1];
    const float* co_A    = (const float*)d_in[3];
    const float* co_B    = (const float*)d_in[4];
    const float* co_C    = (const float*)d_in[5];
    const float* co_D    = (const float*)d_in[6];
    const float* sigma_u = (const float*)d_in[7];
    const float* sigma_v = (const float*)d_in[8];
    float* out = (float*)d_out;

    rsdpf_filter_kernel<<<BATCH, NTHREADS, 0, stream>>>(
        o, P, co_A, co_B, co_C, co_D, sigma_u, sigma_v, out);
}